// QuantumLearnablePositionalEncoding_38482906972554
// MI455X (gfx1250) — compile-verified
//
#include <hip/hip_runtime.h>
#include <hip/hip_bf16.h>

typedef float v2f __attribute__((ext_vector_type(2)));
typedef float v8f __attribute__((ext_vector_type(8)));

#define NQ      11
#define QDIM    2048
#define NLAYERS 2
#define THREADS 128   // 4 wave32

// View of the statevector as S_v[k (0..15), h (0..127)] -> linear amplitude index.
// GATE 0: k = bits 0..3      n = h*16 + k
// GATE 1: k = bits 4..7      n = (h>>4)*256 + k*16 + (h&15)
// GATE 2: k = bits 7..10     n = k*128 + h
template <int GATE>
__device__ __forceinline__ int view_idx(int k, int h) {
  if (GATE == 0)      return (h << 4) | k;
  else if (GATE == 1) return (((h >> 4) << 8)) | (k << 4) | (h & 15);
  else                return (k << 7) | h;
}

// One fused 16-dim gate: D(16x128) = G(16x16) x S_view(16x128), complex f32,
// via chained V_WMMA_F32_16X16X4_F32. Each wave owns 2 of the 8 column tiles;
// a tile writes exactly the columns it read, so in-place update is race-free.
template <int GATE>
__device__ __forceinline__ void apply_gate(float* s_re, float* s_im,
                                           const float* Gr, const float* Gi,
                                           int wave, int mrow, int khalf) {
  // A fragments (G matrix) are tile-independent: load once into registers.
  v2f aR[4], aI[4], nAI[4];
#pragma unroll
  for (int kk = 0; kk < 4; ++kk) {
    int k0 = kk * 4 + khalf * 2;               // even -> 8B-aligned pair
    aR[kk]  = v2f{  Gr[mrow * 16 + k0],  Gr[mrow * 16 + k0 + 1] };
    aI[kk]  = v2f{  Gi[mrow * 16 + k0],  Gi[mrow * 16 + k0 + 1] };
    nAI[kk] = v2f{ -aI[kk].x, -aI[kk].y };
  }
#pragma unroll
  for (int t = 0; t < 2; ++t) {
    const int hcol = ((wave << 1) + t) * 16 + mrow;   // this lane's N column
    v8f accR = {0.f,0.f,0.f,0.f,0.f,0.f,0.f,0.f};
    v8f accI = {0.f,0.f,0.f,0.f,0.f,0.f,0.f,0.f};
#pragma unroll
    for (int kk = 0; kk < 4; ++kk) {
      const int k0 = kk * 4 + khalf * 2;
      const int n0 = view_idx<GATE>(k0,     hcol);
      const int n1 = view_idx<GATE>(k0 + 1, hcol);
      v2f bR = { s_re[n0], s_re[n1] };
      v2f bI = { s_im[n0], s_im[n1] };
      // D_re += Gr*Sr - Gi*Si ; D_im += Gr*Si + Gi*Sr
      accR = __builtin_amdgcn_wmma_f32_16x16x4_f32(false, aR[kk],  false, bR, (short)0, accR, false, false);
      accR = __builtin_amdgcn_wmma_f32_16x16x4_f32(false, nAI[kk], false, bI, (short)0, accR, false, false);
      accI = __builtin_amdgcn_wmma_f32_16x16x4_f32(false, aR[kk],  false, bI, (short)0, accI, false, false);
      accI = __builtin_amdgcn_wmma_f32_16x16x4_f32(false, aI[kk],  false, bR, (short)0, accI, false, false);
    }
    // D tile: vgpr r, lane -> row j = r + 8*khalf, col N = mrow
#pragma unroll
    for (int r = 0; r < 8; ++r) {
      const int j = r + 8 * khalf;
      const int n = view_idx<GATE>(j, hcol);
      s_re[n] = accR[r];
      s_im[n] = accI[r];
    }
  }
}

__global__ __launch_bounds__(THREADS)
void QuantumPQC_wmma_kernel(const float* __restrict__ theta,
                            const int*   __restrict__ positions,
                            float*       __restrict__ out) {
  __shared__ float s_re[QDIM];
  __shared__ float s_im[QDIM];
  __shared__ float g_re[3 * 256];   // three 16x16 complex gate matrices, row-major [j*16+k]
  __shared__ float g_im[3 * 256];
  __shared__ float u_re[NQ * 4];    // per-qubit 2x2 U, [q*4 + i*2 + j]
  __shared__ float u_im[NQ * 4];

  const int tid   = threadIdx.x;
  const int lane  = tid & 31;
  const int wave  = tid >> 5;
  const int mrow  = lane & 15;   // M (A rows) / N (B,D cols)
  const int khalf = lane >> 4;   // selects K pair {0,1} vs {2,3} within a 16x16x4 step

  const int b    = blockIdx.x;
  const int bsrc = positions[b];
  const float* thb = theta + (size_t)bsrc * (NLAYERS * 3 * NQ);

  // |0...0>
  for (int i = tid; i < QDIM; i += THREADS) { s_re[i] = 0.f; s_im[i] = 0.f; }
  if (tid == 0) s_re[0] = 1.f;
  __syncthreads();

  for (int l = 0; l < NLAYERS; ++l) {
    // ---- per-qubit U = Rz(g) * Ry(b) * Rx(a), half angles ----
    if (tid < NQ) {
      const float* t3 = thb + l * (3 * NQ) + 3 * tid;
      float a = 0.5f * t3[0], bb = 0.5f * t3[1], g = 0.5f * t3[2];
      float ca = cosf(a),  sa = sinf(a);
      float cb = cosf(bb), sb = sinf(bb);
      float cg = cosf(g),  sg = sinf(g);
      // M = Ry*Rx:
      float r00 =  cb * ca, i00 =  sb * sa;
      float r01 = -sb * ca, i01 = -cb * sa;
      float r10 =  sb * ca, i10 = -cb * sa;
      float r11 =  cb * ca, i11 = -sb * sa;
      // row0 *= e^{-ig}, row1 *= e^{+ig}
      float* ur = u_re + tid * 4;
      float* ui = u_im + tid * 4;
      ur[0] = cg * r00 + sg * i00;  ui[0] = cg * i00 - sg * r00;
      ur[1] = cg * r01 + sg * i01;  ui[1] = cg * i01 - sg * r01;
      ur[2] = cg * r10 - sg * i10;  ui[2] = cg * i10 + sg * r10;
      ur[3] = cg * r11 - sg * i11;  ui[3] = cg * i11 + sg * r11;
    }
    __syncthreads();

    // ---- build the three 16x16 Kronecker gate matrices ----
    for (int e = tid; e < 768; e += THREADS) {
      int mat = e >> 8, idx = e & 255;
      int j = idx >> 4, k = idx & 15;
      float pr = 1.f, pi = 0.f;
      if (mat < 2) {
        int qb = mat * 4;                       // qubits qb..qb+3, bit q of j/k
#pragma unroll
        for (int q = 0; q < 4; ++q) {
          int ji = (j >> q) & 1, ki = (k >> q) & 1;
          float urv = u_re[(qb + q) * 4 + ji * 2 + ki];
          float uiv = u_im[(qb + q) * 4 + ji * 2 + ki];
          float t = pr * urv - pi * uiv;
          pi = pr * uiv + pi * urv;  pr = t;
        }
      } else {
        // bits 7..10 of the index; identity on bit 7 (slot 0), U8..U10 on slots 1..3
        if ((j & 1) != (k & 1)) { pr = 0.f; pi = 0.f; }
        else {
#pragma unroll
          for (int q = 1; q < 4; ++q) {
            int ji = (j >> q) & 1, ki = (k >> q) & 1;
            float urv = u_re[(7 + q) * 4 + ji * 2 + ki];
            float uiv = u_im[(7 + q) * 4 + ji * 2 + ki];
            float t = pr * urv - pi * uiv;
            pi = pr * uiv + pi * urv;  pr = t;
          }
        }
      }
      g_re[e] = pr;
      g_im[e] = pi;
    }
    __syncthreads();

    // ---- apply the three fused gates (compile-time views => pure shift/or addressing) ----
    apply_gate<0>(s_re, s_im, g_re + 0,   g_im + 0,   wave, mrow, khalf);
    __syncthreads();
    apply_gate<1>(s_re, s_im, g_re + 256, g_im + 256, wave, mrow, khalf);
    __syncthreads();
    apply_gate<2>(s_re, s_im, g_re + 512, g_im + 512, wave, mrow, khalf);
    __syncthreads();

    // ---- CNOT ladder q->q+1 for q=0..9 composes to: final[m] = old[m ^ (m>>1)] ----
    float tr[16], ti[16];
#pragma unroll
    for (int i = 0; i < 16; ++i) {
      int m = tid + i * THREADS;
      int src = m ^ (m >> 1);
      tr[i] = s_re[src];
      ti[i] = s_im[src];
    }
    __syncthreads();
#pragma unroll
    for (int i = 0; i < 16; ++i) {
      int m = tid + i * THREADS;
      s_re[m] = tr[i];
      s_im[m] = ti[i];
    }
    __syncthreads();
  }

  // real part out, coalesced
  float* ob = out + (size_t)b * QDIM;
  for (int i = tid; i < QDIM; i += THREADS) ob[i] = s_re[i];
}

extern "C" void kernel_launch(void* const* d_in, const int* in_sizes, int n_in,
                              void* d_out, int out_size, void* d_ws, size_t ws_size,
                              hipStream_t stream) {
  const float* theta     = (const float*)d_in[0];
  const int*   positions = (const int*)d_in[1];
  float*       out       = (float*)d_out;
  const int B = in_sizes[1];   // number of positions (4096)
  dim3 grid(B), block(THREADS);
  QuantumPQC_wmma_kernel<<<grid, block, 0, stream>>>(theta, positions, out);
}